// GraphModule_80874234184506
// MI455X (gfx1250) — compile-verified
//
#include <hip/hip_runtime.h>

typedef float v2f __attribute__((ext_vector_type(2)));
typedef float v8f __attribute__((ext_vector_type(8)));

#define NN 100000
#define NE 1600000
#define DF 64

__global__ void zero_kernel(float* __restrict__ p, int n) {
    int i = blockIdx.x * blockDim.x + threadIdx.x;
    if (i < n) p[i] = 0.0f;
}

__global__ void degree_kernel(const long long* __restrict__ ei, float* __restrict__ cnt) {
    int e = blockIdx.x * blockDim.x + threadIdx.x;
    if (e < NE) {
        int dst = (int)ei[NE + e];
        atomicAdd(&cnt[dst], 1.0f);
    }
}

// One float4 chunk per (edge, q) pair: 16 lanes cover a full 256B feature row.
__global__ void scatter_kernel(const float* __restrict__ feat, const long long* __restrict__ ei,
                               float* __restrict__ msg) {
    int t = blockIdx.x * blockDim.x + threadIdx.x;
    int e = t >> 4;
    int q = t & 15;
    if (e < NE) {
        int src = (int)ei[e];
        int dst = (int)ei[NE + e];
        const float4 v = *(const float4*)(feat + (size_t)src * DF + q * 4);
        float* mp = msg + (size_t)dst * DF + q * 4;
        atomicAdd(mp + 0, v.x);
        atomicAdd(mp + 1, v.y);
        atomicAdd(mp + 2, v.z);
        atomicAdd(mp + 3, v.w);
    }
}

// out[i,:] = (msg[i,:]/max(cnt[i],1)) @ w_l^T + b_l + xin[i,:] @ w_r^T   (optional ReLU)
// One wave per 16-row block; 4x 16-col tiles; K=64 in steps of 4 via f32 WMMA.
__global__ void __launch_bounds__(256)
sage_combine_kernel(const float* __restrict__ msg, const float* __restrict__ cnt,
                    const float* __restrict__ xin,
                    const float* __restrict__ w_l, const float* __restrict__ b_l,
                    const float* __restrict__ w_r,
                    float* __restrict__ out, int relu) {
    __shared__ float s_wl[DF * 65];   // padded stride 65 -> conflict-free col reads
    __shared__ float s_wr[DF * 65];
    for (int i = threadIdx.x; i < DF * DF; i += 256) {
        int n = i >> 6, k = i & 63;
        s_wl[n * 65 + k] = w_l[i];
        s_wr[n * 65 + k] = w_r[i];
    }
    __syncthreads();

    const int wave = threadIdx.x >> 5;
    const int lane = threadIdx.x & 31;
    const int m    = lane & 15;   // A-row / B-col / C-col within tile
    const int half = lane >> 4;   // K-pair selector for A/B, row-half for C/D
    const int row0 = (blockIdx.x * 8 + wave) * 16;

    // Clamp row for loads (keeps EXEC all-1s for WMMA); stores are guarded.
    int row    = row0 + m;
    int rclamp = row < NN ? row : NN - 1;
    float inv  = 1.0f / fmaxf(cnt[rclamp], 1.0f);
    const float* msrow = msg + (size_t)rclamp * DF;
    const float* xrow  = xin + (size_t)rclamp * DF;

    v8f acc[4] = {v8f{}, v8f{}, v8f{}, v8f{}};

    for (int kb = 0; kb < DF; kb += 4) {
        int k0 = kb + half * 2;
        v2f am, ax;
        am.x = msrow[k0]     * inv;
        am.y = msrow[k0 + 1] * inv;
        ax.x = xrow[k0];
        ax.y = xrow[k0 + 1];
#pragma unroll
        for (int nb = 0; nb < 4; ++nb) {
            int col = nb * 16 + m;
            v2f bl, br;
            bl.x = s_wl[col * 65 + k0];
            bl.y = s_wl[col * 65 + k0 + 1];
            br.x = s_wr[col * 65 + k0];
            br.y = s_wr[col * 65 + k0 + 1];
            // D = A*B + C : 8-arg form (neg_a, A, neg_b, B, c_mod, C, reuse_a, reuse_b)
            acc[nb] = __builtin_amdgcn_wmma_f32_16x16x4_f32(
                false, am, false, bl, (short)0, acc[nb], false, false);
            acc[nb] = __builtin_amdgcn_wmma_f32_16x16x4_f32(
                false, ax, false, br, (short)0, acc[nb], false, false);
        }
    }

#pragma unroll
    for (int nb = 0; nb < 4; ++nb) {
        int col = nb * 16 + m;
        float bias = b_l[col];
#pragma unroll
        for (int r = 0; r < 8; ++r) {
            int orow = row0 + half * 8 + r;
            if (orow < NN) {
                float v = acc[nb][r] + bias;
                if (relu) v = fmaxf(v, 0.0f);
                out[(size_t)orow * DF + col] = v;
            }
        }
    }
}

extern "C" void kernel_launch(void* const* d_in, const int* in_sizes, int n_in,
                              void* d_out, int out_size, void* d_ws, size_t ws_size,
                              hipStream_t stream) {
    const float*     x    = (const float*)d_in[0];
    const long long* ei   = (const long long*)d_in[1];   // int64 [2, E]
    const float*     w_l1 = (const float*)d_in[2];
    const float*     b_l1 = (const float*)d_in[3];
    const float*     w_r1 = (const float*)d_in[4];
    const float*     w_l2 = (const float*)d_in[5];
    const float*     b_l2 = (const float*)d_in[6];
    const float*     w_r2 = (const float*)d_in[7];
    float* out = (float*)d_out;

    float* msg = (float*)d_ws;                 // [NN, DF]
    float* cnt = msg + (size_t)NN * DF;        // [NN]  (contiguous after msg)
    float* h   = cnt + NN;                     // [NN, DF]

    const int threads = 256;
    // zero msg + cnt in one pass (contiguous)
    int zt = NN * DF + NN;
    zero_kernel<<<(zt + threads - 1) / threads, threads, 0, stream>>>(msg, zt);
    degree_kernel<<<(NE + threads - 1) / threads, threads, 0, stream>>>(ei, cnt);

    // Layer 1
    scatter_kernel<<<(NE * 16 + threads - 1) / threads, threads, 0, stream>>>(x, ei, msg);
    sage_combine_kernel<<<(NN + 127) / 128, threads, 0, stream>>>(msg, cnt, x,
                                                                  w_l1, b_l1, w_r1, h, 1);
    // Layer 2
    zero_kernel<<<(NN * DF + threads - 1) / threads, threads, 0, stream>>>(msg, NN * DF);
    scatter_kernel<<<(NE * 16 + threads - 1) / threads, threads, 0, stream>>>(h, ei, msg);
    sage_combine_kernel<<<(NN + 127) / 128, threads, 0, stream>>>(msg, cnt, h,
                                                                  w_l2, b_l2, w_r2, out, 0);
}